// NodeColor_24927990186018
// MI455X (gfx1250) — compile-verified
//
#include <hip/hip_runtime.h>

#define NUM_GRAPHS 128
#define HIDDEN 256
#define EPSV 1e-8f

typedef float v2f __attribute__((ext_vector_type(2)));
typedef float v8f __attribute__((ext_vector_type(8)));

// ---------------------------------------------------------------------------
// Kernel 1: deterministic scatter-mean of node_pos by (sorted) batch id.
// One block per graph; binary-search the segment, strided partial sums,
// fixed-order LDS tree reduction -> bitwise deterministic across replays.
// ctr layout: float4 per graph {cx, cy, cz, count}.
// ---------------------------------------------------------------------------
__global__ __launch_bounds__(256) void centers_kernel(
    const float* __restrict__ pos, const int* __restrict__ batch, int N,
    float* __restrict__ ctr) {
  const int g = blockIdx.x;

  // lower_bound(g)
  int a = 0, b = N;
  while (a < b) { int mid = (a + b) >> 1; if (batch[mid] < g) a = mid + 1; else b = mid; }
  const int start = a;
  // lower_bound(g+1)
  b = N;
  while (a < b) { int mid = (a + b) >> 1; if (batch[mid] < g + 1) a = mid + 1; else b = mid; }
  const int end = a;

  float sx = 0.f, sy = 0.f, sz = 0.f;
  for (int i = start + (int)threadIdx.x; i < end; i += 256) {
    sx += pos[3 * i + 0];
    sy += pos[3 * i + 1];
    sz += pos[3 * i + 2];
  }

  __shared__ float rx[256], ry[256], rz[256];
  rx[threadIdx.x] = sx; ry[threadIdx.x] = sy; rz[threadIdx.x] = sz;
  __syncthreads();
  for (int s = 128; s > 0; s >>= 1) {
    if ((int)threadIdx.x < s) {
      rx[threadIdx.x] += rx[threadIdx.x + s];
      ry[threadIdx.x] += ry[threadIdx.x + s];
      rz[threadIdx.x] += rz[threadIdx.x + s];
    }
    __syncthreads();
  }
  if (threadIdx.x == 0) {
    const int cnt = end - start;
    const float inv = cnt > 0 ? (1.0f / (float)cnt) : 0.0f;
    ctr[4 * g + 0] = rx[0] * inv;
    ctr[4 * g + 1] = ry[0] * inv;
    ctr[4 * g + 2] = rz[0] * inv;
    ctr[4 * g + 3] = (float)cnt;
  }
}

// ---------------------------------------------------------------------------
// Kernel 2: fused  scalar -> silu MLP layer 1 -> GEMM (h @ W2 + b2)
// via V_WMMA_F32_16X16X4_F32 (exact fp32 path on the matrix pipe).
//
// Block = 256 threads (8 waves). Block owns a 16-node M-tile and all 256
// output columns. Each wave computes TWO 16x16 output tiles in a single
// fused K-loop: one A fragment feeds two WMMAs (two accumulators), halving
// LDS traffic vs. per-tile loops.
//
// LDS h tile uses a 260-float row stride: 260 mod 64 == 4, so the 16x4 A
// fragment ds_load_b64 (16 rows at the same k) hits 32 distinct banks.
// ---------------------------------------------------------------------------
#define HROW 260  // padded row stride (floats) for the LDS h tile

__global__ __launch_bounds__(256) void fused_mlp_kernel(
    const float* __restrict__ pos, const int* __restrict__ batch,
    const float* __restrict__ W1, const float* __restrict__ b1,
    const float* __restrict__ W2, const float* __restrict__ b2,
    const float* __restrict__ ctr, float* __restrict__ out, int N) {
  __shared__ float sc[16];
  __shared__ float hLds[16 * HROW];

  const int row0 = blockIdx.x * 16;
  const int tid = threadIdx.x;

  // --- per-node scalar = max(||pos - center[batch]||, eps) (threads 0..15)
  if (tid < 16) {
    int m = row0 + tid;
    int mc = m < N ? m : (N - 1);  // clamp tail rows (stores are masked later)
    int g = batch[mc];
    float dx = pos[3 * mc + 0] - ctr[4 * g + 0];
    float dy = pos[3 * mc + 1] - ctr[4 * g + 1];
    float dz = pos[3 * mc + 2] - ctr[4 * g + 2];
    float r = sqrtf(dx * dx + dy * dy + dz * dz);
    sc[tid] = fmaxf(r, EPSV);
  }
  __syncthreads();

  // --- h[m][k] = silu(scalar[m] * W1[k] + b1[k]) into LDS (k = tid)
  {
    const float w1 = W1[tid];
    const float bb = b1[tid];
#pragma unroll
    for (int m = 0; m < 16; ++m) {
      float x = fmaf(sc[m], w1, bb);
      hLds[m * HROW + tid] = x / (1.0f + __expf(-x));
    }
  }
  __syncthreads();

  // --- WMMA GEMM: out[16, 256] = h[16, 256] @ W2[256, 256] + b2
  const int wave = tid >> 5;
  const int lane = tid & 31;
  const int half = lane >> 4;  // 0: K rows k+0/k+1   1: K rows k+2/k+3
  const int l16 = lane & 15;

  // A fragment source: lane reads h[row=l16][k + 2*half .. +1] (ds_load_b64)
  const float* __restrict__ hRow = &hLds[l16 * HROW + 2 * half];

  // Two adjacent 16-column tiles per wave: columns [wave*32, wave*32+32)
  const int n = wave * 32 + l16;  // column for tile 0 (tile 1 = n + 16)
  const float* __restrict__ pB0 = W2 + (2 * half) * HIDDEN + n;
  const float* __restrict__ pB1 = pB0 + 16;

  v8f acc0 = {};
  v8f acc1 = {};
#pragma unroll 8
  for (int k = 0; k < HIDDEN; k += 4) {
    v2f a;
    {
      float2 av = *(const float2*)(hRow + k);  // 8B-aligned (k even, HROW*4 % 8 == 0)
      a.x = av.x;
      a.y = av.y;
    }
    v2f bf0, bf1;
    bf0.x = pB0[k * HIDDEN];           // W2[k + 2*half    ][n]
    bf0.y = pB0[k * HIDDEN + HIDDEN];  // W2[k + 2*half + 1][n]
    bf1.x = pB1[k * HIDDEN];           // W2[k + 2*half    ][n + 16]
    bf1.y = pB1[k * HIDDEN + HIDDEN];  // W2[k + 2*half + 1][n + 16]
    acc0 = __builtin_amdgcn_wmma_f32_16x16x4_f32(false, a, false, bf0,
                                                 (short)0, acc0, false, false);
    acc1 = __builtin_amdgcn_wmma_f32_16x16x4_f32(false, a, false, bf1,
                                                 (short)0, acc1, false, false);
  }

  // --- epilogue: C/D layout VGPR r -> row r (lanes 0-15) / row 8+r
  // (lanes 16-31). Block-uniform full/tail branch: N % 16 == 0 in practice,
  // so the unguarded fast path avoids per-row exec-mask gymnastics.
  const float bias0 = b2[n];
  const float bias1 = b2[n + 16];
  const int mbase = row0 + half * 8;

  if (row0 + 16 <= N) {  // uniform across the block: one branch, no exec masks
#pragma unroll
    for (int r = 0; r < 8; ++r) {
      const int m = mbase + r;
      __builtin_nontemporal_store(acc0[r] + bias0, &out[m * HIDDEN + n]);
      __builtin_nontemporal_store(acc1[r] + bias1, &out[m * HIDDEN + n + 16]);
    }
  } else {
#pragma unroll
    for (int r = 0; r < 8; ++r) {
      const int m = mbase + r;
      if (m < N) {
        __builtin_nontemporal_store(acc0[r] + bias0, &out[m * HIDDEN + n]);
        __builtin_nontemporal_store(acc1[r] + bias1, &out[m * HIDDEN + n + 16]);
      }
    }
  }
}

// ---------------------------------------------------------------------------
// Launch
// ---------------------------------------------------------------------------
extern "C" void kernel_launch(void* const* d_in, const int* in_sizes, int n_in,
                              void* d_out, int out_size, void* d_ws, size_t ws_size,
                              hipStream_t stream) {
  // setup_inputs order: node_feat, node_pos, batch, W1, b1, W2, b2
  const float* node_pos = (const float*)d_in[1];
  const int* batch = (const int*)d_in[2];
  const float* W1 = (const float*)d_in[3];
  const float* b1 = (const float*)d_in[4];
  const float* W2 = (const float*)d_in[5];
  const float* b2 = (const float*)d_in[6];
  float* out = (float*)d_out;
  float* ctr = (float*)d_ws;  // float4 per graph: {cx, cy, cz, count}

  const int N = in_sizes[2];  // batch has one entry per node

  centers_kernel<<<NUM_GRAPHS, 256, 0, stream>>>(node_pos, batch, N, ctr);

  const int mtiles = (N + 15) / 16;
  fused_mlp_kernel<<<mtiles, 256, 0, stream>>>(node_pos, batch, W1, b1, W2, b2,
                                               ctr, out, N);
}